// ModuleGIN_9122510537161
// MI455X (gfx1250) — compile-verified
//
#include <hip/hip_runtime.h>
#include <hip/hip_bf16.h>

// ---------------------------------------------------------------------------
// GIN 2-layer forward on MI455X (gfx1250, wave32).
// Layer: h = relu( relu( BN( (x + scatter_sum(x)) @ W1 + b1 ) ) @ W2 + b2 )
// - f32 WMMA (v_wmma_f32_16x16x4_f32) for both GEMMs
// - BN batch statistics fused into GEMM1 (LDS ds_add_f32 + global atomics)
// - Weight tile staged to LDS via the Tensor Data Mover (tensor_load_to_lds)
// - 64 rows per block, B fragments held in registers across 4 M-tiles
// ---------------------------------------------------------------------------

#define N_NODES 50000
#define N_EDGES 800000
#define DIM     64
#define BN_EPS  1e-5f

typedef __attribute__((ext_vector_type(2))) float v2f;
typedef __attribute__((ext_vector_type(8))) float v8f;
typedef __attribute__((ext_vector_type(4))) unsigned int v4u;
typedef __attribute__((ext_vector_type(4))) int v4i;
typedef __attribute__((ext_vector_type(8))) int v8i;

#if defined(__has_builtin)
#if __has_builtin(__builtin_amdgcn_tensor_load_to_lds) && \
    __has_builtin(__builtin_amdgcn_s_wait_tensorcnt)
#define HAVE_TDM 1
#endif
#endif
#ifndef HAVE_TDM
#define HAVE_TDM 0
#endif

// Issue a TDM load of a 64x64 f32 tile (row-major, stride 64) into LDS.
// Must be executed by a single wave; wave-uniform arguments.
// This toolchain exposes the 6-arg builtin:
//   (uint32x4 g0, int32x8 g1, int32x4 g2, int32x4 g3, int32x8 g4, i32 cpol)
__device__ __forceinline__ void tdm_load_w(const float* gptr, void* lds) {
#if HAVE_TDM
  const unsigned lds_off = (unsigned)(unsigned long long)(size_t)lds;  // low 32b of flat LDS addr = offset
  const unsigned long long ga = (unsigned long long)(size_t)gptr;

  v4u g0;
  g0[0] = 1u;                                           // count=1, user descriptor
  g0[1] = lds_off;                                      // lds_addr
  g0[2] = (unsigned)(ga & 0xFFFFFFFFu);                 // global_addr[31:0]
  g0[3] = (unsigned)((ga >> 32) & 0x01FFFFFFu)          // global_addr[56:32]
          | (2u << 30);                                 // type=2 ("image")

  v8i g1;
  g1[0] = (int)(2u << 16);        // wg_mask=0, data_size=2 (4 bytes)
  g1[1] = (int)(64u << 16);       // tensor_dim0[15:0] in [31:16]
  g1[2] = (int)(64u << 16);       // tensor_dim0 hi=0; tensor_dim1[15:0] in [31:16]
  g1[3] = (int)(64u << 16);       // tensor_dim1 hi=0; tile_dim0=64 in [31:16]
  g1[4] = 64;                     // tile_dim1=64; tile_dim2=0
  g1[5] = 64;                     // tensor_dim0_stride low 32 = 64
  g1[6] = (int)(4096u << 16);     // stride0 hi=0; tensor_dim1_stride[15:0]=4096
  g1[7] = 0;                      // stride1 hi
  v4i z4 = {0, 0, 0, 0};
  v8i z8 = {0, 0, 0, 0, 0, 0, 0, 0};
  __builtin_amdgcn_tensor_load_to_lds(g0, g1, z4, z4, z8, 0);
#else
  (void)gptr; (void)lds;
#endif
}

// -------------------------------------------------------------------------
// agg = x (covers the "+ x_i" self term), zero the stats accumulators.
// -------------------------------------------------------------------------
__global__ __launch_bounds__(256) void gin_init(const float* __restrict__ x,
                                                float* __restrict__ agg,
                                                float* __restrict__ stats) {
  const long gid = (long)blockIdx.x * blockDim.x + threadIdx.x;
  const long n4 = (long)N_NODES * DIM / 4;
  if (gid < n4) {
    ((float4*)agg)[gid] = ((const float4*)x)[gid];
  }
  if (gid < 128) stats[gid] = 0.0f;
}

// -------------------------------------------------------------------------
// Edge scatter: agg[dst] += x[src].  16 threads per edge, float4 per thread.
// Random 256B gathers use full cachelines; f32 atomics resolve in L2
// (x and agg both fit in the 192MB L2, so this runs at L2 atomic rate).
// -------------------------------------------------------------------------
__global__ __launch_bounds__(256) void gin_scatter(const float* __restrict__ x,
                                                   const int* __restrict__ ei,
                                                   float* __restrict__ agg) {
  const long gid = (long)blockIdx.x * blockDim.x + threadIdx.x;
  if (gid >= (long)N_EDGES * 16) return;
  const int e = (int)(gid >> 4);
  const int f = (int)(gid & 15) << 2;
  const int src = ei[e];
  const int dst = ei[N_EDGES + e];
  const float4 v = *(const float4*)(x + (long)src * DIM + f);
  float* p = agg + (long)dst * DIM + f;
  atomicAdd(p + 0, v.x);
  atomicAdd(p + 1, v.y);
  atomicAdd(p + 2, v.z);
  atomicAdd(p + 3, v.w);
}

// -------------------------------------------------------------------------
// GEMM1: Hpre = agg @ W1 + b1, accumulating per-column sum/sumsq into
// stats[0:64] / stats[64:128].  Block: 128 threads = 4 wave32, 64 rows
// (4 M-tiles); wave w owns output columns [16w,16w+16); B fragments for
// the whole K=64 are kept in registers across the 4 M-tiles.
// -------------------------------------------------------------------------
__global__ __launch_bounds__(128) void gin_gemm1(const float* __restrict__ A,
                                                 const float* __restrict__ W,
                                                 const float* __restrict__ bias,
                                                 float* __restrict__ Hpre,
                                                 float* __restrict__ stats) {
  __shared__ float Alds[64 * DIM];   // 16 KB
  __shared__ float Wlds[DIM * DIM];  // 16 KB
  __shared__ float sumLds[DIM];
  __shared__ float sqLds[DIM];

  const int tid = threadIdx.x;
  const int row0 = blockIdx.x * 64;
  const int ntiles = min(4, (N_NODES - row0) / 16);  // uniform per block

  if (tid < 32) tdm_load_w(W, Wlds);  // wave 0: DMA weight tile into LDS
#if !HAVE_TDM
  for (int i = tid; i < DIM * DIM / 4; i += 128)
    ((float4*)Wlds)[i] = ((const float4*)W)[i];
#endif

  // Threads stage the A tile (overlaps the TDM transfer).
  for (int i = tid; i < ntiles * 16 * DIM / 4; i += 128)
    ((float4*)Alds)[i] = ((const float4*)(A + (long)row0 * DIM))[i];
  if (tid < DIM) { sumLds[tid] = 0.0f; sqLds[tid] = 0.0f; }

#if HAVE_TDM
  if (tid < 32) __builtin_amdgcn_s_wait_tensorcnt(0);
#endif
  __syncthreads();

  const int wave = tid >> 5;
  const int lane = tid & 31;
  const int m = lane & 15;            // row in tile (A frag) / col id (B,C frag)
  const int khalf = (lane >> 4) * 2;  // K sub-offset for this half-wave
  const int n = (wave << 4) + m;      // output column owned by this lane
  const float bval = bias[n];
  const int mbase = (lane >> 4) * 8;

  // B fragments for all 16 K-steps, register resident.
  v2f bf[16];
#pragma unroll
  for (int k4 = 0; k4 < 16; ++k4) {
    bf[k4].x = Wlds[(k4 * 4 + khalf) * DIM + n];
    bf[k4].y = Wlds[(k4 * 4 + khalf + 1) * DIM + n];
  }

  for (int t = 0; t < ntiles; ++t) {
    const float* at = Alds + (t * 16 + m) * DIM;
    v8f c = {};
#pragma unroll
    for (int k4 = 0; k4 < 16; ++k4) {
      v2f a;
      a.x = at[k4 * 4 + khalf];
      a.y = at[k4 * 4 + khalf + 1];
      c = __builtin_amdgcn_wmma_f32_16x16x4_f32(false, a, false, bf[k4],
                                                (short)0, c, false, false);
    }
    // bias + BN partial stats + store (8 rows per lane).
    float s = 0.0f, sq = 0.0f;
    const int rbase = row0 + t * 16 + mbase;
#pragma unroll
    for (int r = 0; r < 8; ++r) {
      const float v = c[r] + bval;
      s += v;
      sq += v * v;
      Hpre[(long)(rbase + r) * DIM + n] = v;
    }
    atomicAdd(&sumLds[n], s);  // ds_add_f32
    atomicAdd(&sqLds[n], sq);  // ds_add_f32
  }
  __syncthreads();
  if (tid < DIM) {
    atomicAdd(&stats[tid], sumLds[tid]);
    atomicAdd(&stats[DIM + tid], sqLds[tid]);
  }
}

// -------------------------------------------------------------------------
// BN finalize: stats[128:192] = scale = g * rsqrt(var + eps),
//              stats[192:256] = shift = beta - mean * scale.
// -------------------------------------------------------------------------
__global__ __launch_bounds__(64) void gin_bnfin(float* __restrict__ stats,
                                                const float* __restrict__ g,
                                                const float* __restrict__ beta) {
  const int i = threadIdx.x;
  const float inv_n = 1.0f / (float)N_NODES;
  const float mean = stats[i] * inv_n;
  const float var = stats[DIM + i] * inv_n - mean * mean;
  const float a = g[i] * rsqrtf(var + BN_EPS);
  stats[128 + i] = a;
  stats[192 + i] = beta[i] - mean * a;
}

// -------------------------------------------------------------------------
// GEMM2: out = relu( relu(scale*Hpre + shift) @ W2 + b2 ).
// BN affine + ReLU fused into the A-tile staging; W2 via TDM.
// -------------------------------------------------------------------------
__global__ __launch_bounds__(128) void gin_gemm2(const float* __restrict__ Hpre,
                                                 const float* __restrict__ stats,
                                                 const float* __restrict__ W,
                                                 const float* __restrict__ bias,
                                                 float* __restrict__ out) {
  __shared__ float Alds[64 * DIM];
  __shared__ float Wlds[DIM * DIM];
  __shared__ float sc[DIM];
  __shared__ float sh[DIM];

  const int tid = threadIdx.x;
  const int row0 = blockIdx.x * 64;
  const int ntiles = min(4, (N_NODES - row0) / 16);

  if (tid < 32) tdm_load_w(W, Wlds);
#if !HAVE_TDM
  for (int i = tid; i < DIM * DIM / 4; i += 128)
    ((float4*)Wlds)[i] = ((const float4*)W)[i];
#endif

  if (tid < DIM) {
    sc[tid] = stats[128 + tid];
    sh[tid] = stats[192 + tid];
  }
  __syncthreads();  // sc/sh visible before the A-tile transform

  for (int i = tid; i < ntiles * 16 * DIM / 4; i += 128) {
    float4 v = ((const float4*)(Hpre + (long)row0 * DIM))[i];
    const int kb = (i & (DIM / 4 - 1)) * 4;
    v.x = fmaxf(fmaf(sc[kb + 0], v.x, sh[kb + 0]), 0.0f);
    v.y = fmaxf(fmaf(sc[kb + 1], v.y, sh[kb + 1]), 0.0f);
    v.z = fmaxf(fmaf(sc[kb + 2], v.z, sh[kb + 2]), 0.0f);
    v.w = fmaxf(fmaf(sc[kb + 3], v.w, sh[kb + 3]), 0.0f);
    ((float4*)Alds)[i] = v;
  }

#if HAVE_TDM
  if (tid < 32) __builtin_amdgcn_s_wait_tensorcnt(0);
#endif
  __syncthreads();

  const int wave = tid >> 5;
  const int lane = tid & 31;
  const int m = lane & 15;
  const int khalf = (lane >> 4) * 2;
  const int n = (wave << 4) + m;
  const float bval = bias[n];
  const int mbase = (lane >> 4) * 8;

  v2f bf[16];
#pragma unroll
  for (int k4 = 0; k4 < 16; ++k4) {
    bf[k4].x = Wlds[(k4 * 4 + khalf) * DIM + n];
    bf[k4].y = Wlds[(k4 * 4 + khalf + 1) * DIM + n];
  }

  for (int t = 0; t < ntiles; ++t) {
    const float* at = Alds + (t * 16 + m) * DIM;
    v8f c = {};
#pragma unroll
    for (int k4 = 0; k4 < 16; ++k4) {
      v2f a;
      a.x = at[k4 * 4 + khalf];
      a.y = at[k4 * 4 + khalf + 1];
      c = __builtin_amdgcn_wmma_f32_16x16x4_f32(false, a, false, bf[k4],
                                                (short)0, c, false, false);
    }
    const int rbase = row0 + t * 16 + mbase;
#pragma unroll
    for (int r = 0; r < 8; ++r)
      out[(long)(rbase + r) * DIM + n] = fmaxf(c[r] + bval, 0.0f);
  }
}

// ---------------------------------------------------------------------------
// Launcher. Inputs (setup_inputs order):
//  0:x 1:edge_index 2:W1_0 3:b1_0 4:g_0 5:beta_0 6:W2_0 7:b2_0
//  8:W1_1 9:b1_1 10:g_1 11:beta_1 12:W2_1 13:b2_1
// d_out = [h1 (N*D) | h2 (N*D)].
// ---------------------------------------------------------------------------
extern "C" void kernel_launch(void* const* d_in, const int* in_sizes, int n_in,
                              void* d_out, int out_size, void* d_ws, size_t ws_size,
                              hipStream_t stream) {
  const float* x  = (const float*)d_in[0];
  const int*   ei = (const int*)d_in[1];

  const float* W1[2]   = {(const float*)d_in[2],  (const float*)d_in[8]};
  const float* b1[2]   = {(const float*)d_in[3],  (const float*)d_in[9]};
  const float* g[2]    = {(const float*)d_in[4],  (const float*)d_in[10]};
  const float* beta[2] = {(const float*)d_in[5],  (const float*)d_in[11]};
  const float* W2[2]   = {(const float*)d_in[6],  (const float*)d_in[12]};
  const float* b2[2]   = {(const float*)d_in[7],  (const float*)d_in[13]};

  float* agg   = (float*)d_ws;                // N*D f32
  float* hpre  = agg + (long)N_NODES * DIM;   // N*D f32
  float* stats = hpre + (long)N_NODES * DIM;  // 256 f32

  float* h1 = (float*)d_out;
  float* h2 = h1 + (long)N_NODES * DIM;

  const int initGrid    = (int)(((long)N_NODES * DIM / 4 + 255) / 256);
  const int scatterGrid = (int)(((long)N_EDGES * 16 + 255) / 256);
  const int gemmGrid    = (N_NODES / 16 + 3) / 4;  // 782 blocks of 64 rows

  const float* layer_in = x;
  float*       layer_out[2] = {h1, h2};

  for (int l = 0; l < 2; ++l) {
    gin_init<<<initGrid, 256, 0, stream>>>(layer_in, agg, stats);
    gin_scatter<<<scatterGrid, 256, 0, stream>>>(layer_in, ei, agg);
    gin_gemm1<<<gemmGrid, 128, 0, stream>>>(agg, W1[l], b1[l], hpre, stats);
    gin_bnfin<<<1, 64, 0, stream>>>(stats, g[l], beta[l]);
    gin_gemm2<<<gemmGrid, 128, 0, stream>>>(hpre, stats, W2[l], b2[l], layer_out[l]);
    layer_in = layer_out[l];
  }
}